// FADELite_21449066676349
// MI455X (gfx1250) — compile-verified
//
#include <hip/hip_runtime.h>
#include <math.h>

// FADE-Lite for MI455X (gfx1250, wave32).
// k1/k2: fp32 WMMA GEMMs (V_WMMA_F32_16X16X4_F32) for the 1x1 convs.
// k3: depthwise3x3 + up2(depthwise3x3) + softmax -> CARAFE kernels.
// k4: LDS-tiled CARAFE + gated blend (memory-bound stage, coalesced I/O).

typedef __attribute__((ext_vector_type(2))) float v2f;
typedef __attribute__((ext_vector_type(8))) float v8f;

#define C_IN   256
#define P_HI   16384   /* 128*128 */
#define P_LO   4096    /* 64*64   */
#define KK     25

// ---------------------------------------------------------------------------
// Kernel 1: enc[b,n,p] = sum_c en[b,c,p] * W_ce[n,c] + b_ce[n]
// M = B*P_HI pixels, N = 25 (padded to 32), K = 256.  One 16-row M-tile/wave.
// ---------------------------------------------------------------------------
__global__ __launch_bounds__(256) void k1_enc_gemm(
    const float* __restrict__ X, const float* __restrict__ Wce,
    const float* __restrict__ bce, float* __restrict__ enc)
{
    __shared__ float Wl[32 * 257];           // [n][c], padded stride 257
    const int tid = threadIdx.x;
    for (int i = tid; i < 32 * 256; i += 256) {
        int n = i >> 8, c = i & 255;
        Wl[n * 257 + c] = (n < KK) ? Wce[n * 256 + c] : 0.0f;
    }
    __syncthreads();

    const int wave = tid >> 5, lane = tid & 31;
    const int kh = lane >> 4;                // lane half: 0 or 1
    const int lr = lane & 15;
    const int koff = kh * 2;                 // K offset within 4-wide slab
    const int mtile = blockIdx.x * 8 + wave; // 0..4095
    const int g = mtile * 16 + lr;           // global pixel row for A
    const int b = g >> 14;                   // P_HI = 2^14
    const int p = g & (P_HI - 1);
    const float* Xb = X + ((size_t)(b * C_IN)) * P_HI + p;

    v8f acc0 = {}; v8f acc1 = {};
    for (int c0 = 0; c0 < C_IN; c0 += 4) {
        v2f a;  a.x  = Xb[(size_t)(c0 + koff) * P_HI];
                a.y  = Xb[(size_t)(c0 + koff + 1) * P_HI];
        v2f b0; b0.x = Wl[lr * 257 + c0 + koff];
                b0.y = Wl[lr * 257 + c0 + koff + 1];
        v2f b1; b1.x = Wl[(16 + lr) * 257 + c0 + koff];
                b1.y = Wl[(16 + lr) * 257 + c0 + koff + 1];
        acc0 = __builtin_amdgcn_wmma_f32_16x16x4_f32(false, a, false, b0,
                                                     (short)0, acc0, false, false);
        acc1 = __builtin_amdgcn_wmma_f32_16x16x4_f32(false, a, false, b1,
                                                     (short)0, acc1, false, false);
    }

    // C/D layout: lane L holds (M = r + 8*(L>=16), N = L&15) in VGPR r.
    const int pout = (mtile * 16 + kh * 8) & (P_HI - 1);
    {
        int n = lr;                          // 0..15, always valid
        float bias = bce[n];
        float* dst = enc + ((size_t)(b * KK + n)) * P_HI + pout;
        #pragma unroll
        for (int r = 0; r < 8; ++r) dst[r] = acc0[r] + bias;
    }
    {
        int n = 16 + lr;
        if (n < KK) {
            float bias = bce[n];
            float* dst = enc + ((size_t)(b * KK + n)) * P_HI + pout;
            #pragma unroll
            for (int r = 0; r < 8; ++r) dst[r] = acc1[r] + bias;
        }
    }
}

// ---------------------------------------------------------------------------
// Kernel 2: dec[b,n,p] = sum_c de[b,c,p]*W_cd[n,c]   (n = 0..24)
//           gate[b,p]  = sigmoid(sum_c de*W_gate + b_gate)   (packed as n=25)
// ---------------------------------------------------------------------------
__global__ __launch_bounds__(256) void k2_dec_gemm(
    const float* __restrict__ X, const float* __restrict__ Wcd,
    const float* __restrict__ Wg, const float* __restrict__ bg,
    float* __restrict__ dec, float* __restrict__ gate)
{
    __shared__ float Wl[32 * 257];
    const int tid = threadIdx.x;
    for (int i = tid; i < 32 * 256; i += 256) {
        int n = i >> 8, c = i & 255;
        float v = 0.0f;
        if (n < KK) v = Wcd[n * 256 + c];
        else if (n == KK) v = Wg[c];
        Wl[n * 257 + c] = v;
    }
    __syncthreads();

    const int wave = tid >> 5, lane = tid & 31;
    const int kh = lane >> 4, lr = lane & 15, koff = kh * 2;
    const int mtile = blockIdx.x * 8 + wave; // 0..1023
    const int g = mtile * 16 + lr;
    const int b = g >> 12;                   // P_LO = 2^12
    const int p = g & (P_LO - 1);
    const float* Xb = X + ((size_t)(b * C_IN)) * P_LO + p;

    v8f acc0 = {}; v8f acc1 = {};
    for (int c0 = 0; c0 < C_IN; c0 += 4) {
        v2f a;  a.x  = Xb[(size_t)(c0 + koff) * P_LO];
                a.y  = Xb[(size_t)(c0 + koff + 1) * P_LO];
        v2f b0; b0.x = Wl[lr * 257 + c0 + koff];
                b0.y = Wl[lr * 257 + c0 + koff + 1];
        v2f b1; b1.x = Wl[(16 + lr) * 257 + c0 + koff];
                b1.y = Wl[(16 + lr) * 257 + c0 + koff + 1];
        acc0 = __builtin_amdgcn_wmma_f32_16x16x4_f32(false, a, false, b0,
                                                     (short)0, acc0, false, false);
        acc1 = __builtin_amdgcn_wmma_f32_16x16x4_f32(false, a, false, b1,
                                                     (short)0, acc1, false, false);
    }

    const int pout = (mtile * 16 + kh * 8) & (P_LO - 1);
    {
        int n = lr;
        float* dst = dec + ((size_t)(b * KK + n)) * P_LO + pout;
        #pragma unroll
        for (int r = 0; r < 8; ++r) dst[r] = acc0[r];
    }
    {
        int n = 16 + lr;
        if (n < KK) {
            float* dst = dec + ((size_t)(b * KK + n)) * P_LO + pout;
            #pragma unroll
            for (int r = 0; r < 8; ++r) dst[r] = acc1[r];
        } else if (n == KK) {
            float bias = bg[0];
            float* dst = gate + (size_t)b * P_LO + pout;
            #pragma unroll
            for (int r = 0; r < 8; ++r) {
                float v = acc1[r] + bias;
                dst[r] = 1.0f / (1.0f + __expf(-v));
            }
        }
    }
}

// ---------------------------------------------------------------------------
// Kernel 3: raw = dw3x3(enc) + up2(dw3x3(dec)) ; kernels = softmax_k(raw)
// One thread per hi-res pixel (b,y,x).
// ---------------------------------------------------------------------------
__global__ __launch_bounds__(256) void k3_kernels(
    const float* __restrict__ enc, const float* __restrict__ dec,
    const float* __restrict__ Wdw, const float* __restrict__ bdw,
    float* __restrict__ kern)
{
    __shared__ float sW[KK * 9];
    __shared__ float sb[KK];
    const int tid = threadIdx.x;
    if (tid < KK * 9) sW[tid] = Wdw[tid];
    if (tid < KK)     sb[tid] = bdw[tid];
    __syncthreads();

    const int idx = blockIdx.x * 256 + tid;  // 0..65535
    const int b = idx >> 14;
    const int y = (idx >> 7) & 127;
    const int x = idx & 127;
    const int h = y >> 1, w = x >> 1;
    const float* eb = enc + (size_t)b * KK * P_HI;
    const float* db = dec + (size_t)b * KK * P_LO;

    float raw[KK];
    float mx = -3.0e38f;
    #pragma unroll
    for (int k = 0; k < KK; ++k) {
        float acc = 2.0f * sb[k];            // bias added in both depthwise convs
        #pragma unroll
        for (int i = 0; i < 3; ++i) {
            #pragma unroll
            for (int j = 0; j < 3; ++j) {
                float wv = sW[k * 9 + i * 3 + j];
                int yy = y + i - 1, xx = x + j - 1;
                if (yy >= 0 && yy < 128 && xx >= 0 && xx < 128)
                    acc += wv * eb[(size_t)k * P_HI + yy * 128 + xx];
                int hh = h + i - 1, ww = w + j - 1;
                if (hh >= 0 && hh < 64 && ww >= 0 && ww < 64)
                    acc += wv * db[(size_t)k * P_LO + hh * 64 + ww];
            }
        }
        raw[k] = acc;
        mx = fmaxf(mx, acc);
    }
    float sum = 0.0f;
    #pragma unroll
    for (int k = 0; k < KK; ++k) { raw[k] = __expf(raw[k] - mx); sum += raw[k]; }
    float inv = 1.0f / sum;
    float* kb = kern + (size_t)b * KK * P_HI + y * 128 + x;
    #pragma unroll
    for (int k = 0; k < KK; ++k) kb[(size_t)k * P_HI] = raw[k] * inv;
}

// ---------------------------------------------------------------------------
// Kernel 4: out = gate*en + (1-gate)*CARAFE(de, kernels)
// Block: 8 channels x (4x16 source tile -> 8x32 output tile).
// ---------------------------------------------------------------------------
__global__ __launch_bounds__(256) void k4_carafe(
    const float* __restrict__ en, const float* __restrict__ de,
    const float* __restrict__ kern, const float* __restrict__ gate,
    float* __restrict__ out)
{
    __shared__ float sde[8][8][20];     // [c_local][h_local+dy][w_local+dx]
    __shared__ float sker[KK][8][32];   // [k][py][px]

    const int tid = threadIdx.x;
    const int bx = blockIdx.x;          // 0..63
    const int tw = bx & 3, th = bx >> 2;
    const int h0 = th * 4, w0 = tw * 16;
    const int c0 = blockIdx.y * 8;
    const int b  = blockIdx.z;

    // de patch (zero-padded halo of 2)
    for (int i = tid; i < 8 * 8 * 20; i += 256) {
        int cl = i / 160; int r = i - cl * 160;
        int ph = r / 20,  pw = r - ph * 20;
        int hh = h0 + ph - 2, ww = w0 + pw - 2;
        float v = 0.0f;
        if (hh >= 0 && hh < 64 && ww >= 0 && ww < 64)
            v = de[((size_t)(b * C_IN + c0 + cl)) * P_LO + hh * 64 + ww];
        sde[cl][ph][pw] = v;
    }
    // softmaxed kernels for this output tile
    const float* kb = kern + (size_t)b * KK * P_HI;
    for (int i = tid; i < KK * 256; i += 256) {
        int k = i >> 8; int r = i & 255;
        int py = r >> 5, px = r & 31;
        sker[k][py][px] = kb[(size_t)k * P_HI + (2 * h0 + py) * 128 + (2 * w0 + px)];
    }
    __syncthreads();

    const int px = tid & 31, cl = tid >> 5;
    const int x  = 2 * w0 + px;
    const int wl = px >> 1;
    const size_t cbase = ((size_t)(b * C_IN + c0 + cl)) * P_HI;
    const float* gb = gate + (size_t)b * P_LO;
    __builtin_prefetch(&en[cbase + (2 * h0) * 128 + x], 0, 1);
    __builtin_prefetch(&en[cbase + (2 * h0 + 4) * 128 + x], 0, 1);

    #pragma unroll
    for (int py = 0; py < 8; ++py) {
        const int y = 2 * h0 + py;
        const int hl = py >> 1;
        float acc = 0.0f;
        #pragma unroll
        for (int k = 0; k < KK; ++k) {
            const int dy = k / 5, dx = k - dy * 5;
            acc += sker[k][py][px] * sde[cl][hl + dy][wl + dx];
        }
        float g = gb[(h0 + hl) * 64 + (w0 + wl)];
        float e = en[cbase + y * 128 + x];
        out[cbase + y * 128 + x] = g * e + (1.0f - g) * acc;
    }
}

// ---------------------------------------------------------------------------
extern "C" void kernel_launch(void* const* d_in, const int* in_sizes, int n_in,
                              void* d_out, int out_size, void* d_ws, size_t ws_size,
                              hipStream_t stream) {
    const float* en     = (const float*)d_in[0];
    const float* de     = (const float*)d_in[1];
    const float* W_gate = (const float*)d_in[2];
    const float* b_gate = (const float*)d_in[3];
    const float* W_ce   = (const float*)d_in[4];
    const float* b_ce   = (const float*)d_in[5];
    const float* W_cd   = (const float*)d_in[6];
    const float* W_dw   = (const float*)d_in[7];
    const float* b_dw   = (const float*)d_in[8];
    float* out = (float*)d_out;

    // workspace carve-up (floats): enc 4*25*16384, kern 4*25*16384,
    // dec 4*25*4096, gate 4*4096  => ~14.8 MB total
    float* enc  = (float*)d_ws;
    float* kern = enc  + (size_t)4 * KK * P_HI;
    float* dec  = kern + (size_t)4 * KK * P_HI;
    float* gate = dec  + (size_t)4 * KK * P_LO;

    k1_enc_gemm<<<512, 256, 0, stream>>>(en, W_ce, b_ce, enc);
    k2_dec_gemm<<<128, 256, 0, stream>>>(de, W_cd, W_gate, b_gate, dec, gate);
    k3_kernels<<<256, 256, 0, stream>>>(enc, dec, W_dw, b_dw, kern);
    k4_carafe<<<dim3(64, 32, 4), 256, 0, stream>>>(en, de, kern, gate, out);
}